// BasenetFgnnMeanfield_1305670058142
// MI455X (gfx1250) — compile-verified
//
#include <hip/hip_runtime.h>
#include <hip/hip_bf16.h>

// Problem constants (N = 15 persons)
#define NPER   15
#define NVAR   120    // variable nodes (15 persons + 105 pair z-nodes)
#define NG     680    // total graph nodes
#define DIM    128
#define EDIM   16
#define ZBASE  15     // z-node row offset
#define HBASE  120    // h-factor row offset
#define GBASE  225    // g-factor row offset
#define DEG    14

typedef float v2f __attribute__((ext_vector_type(2)));
typedef float v8f __attribute__((ext_vector_type(8)));

// ---- combinatorial rank helpers (N=15) ----
__device__ __forceinline__ int pair_rank(int u, int v) {
    // rank of pair (u,v), u<v, in lexicographic order
    return u * 15 - ((u * (u + 1)) >> 1) + (v - u - 1);
}
__device__ __forceinline__ void decode_pair(int p, int& u, int& v) {
    int uu = 0;
    while (p >= 14 - uu) { p -= 14 - uu; ++uu; }
    u = uu; v = uu + 1 + p;
}
__device__ __forceinline__ void decode_tri(int t, int& i, int& j, int& k) {
    int ii = 0;
    while (t >= (((14 - ii) * (13 - ii)) >> 1)) { t -= ((14 - ii) * (13 - ii)) >> 1; ++ii; }
    int jj = ii + 1;
    while (t >= 14 - jj) { t -= 14 - jj; ++jj; }
    i = ii; j = jj; k = jj + 1 + t;
}
__device__ __forceinline__ int tri_rank(int i, int j, int k) {
    int A  = 455 - ((15 - i) * (14 - i) * (13 - i)) / 6;  // triples with first < i
    int m  = j - i - 1;
    int Bs = m * (14 - i) - ((m * (m + 1)) >> 1);         // triples with first==i, second<j
    return A + Bs + (k - j - 1);
}

// =====================================================================
// Kernel A: fused nff construction + GEMM  proj = nff @ [W_self | W_nbr]
// One wave (32 lanes) per 16-row M tile; N = 32 (two 16-wide WMMA tiles).
// Branchless A-operand blend keeps EXEC all-ones through the whole loop
// (WMMA requirement) and keeps the inner loop free of exec save/restore.
// =====================================================================
#define LSTRIDE 132   // padded LDS row stride (floats) to avoid bank conflicts

__global__ __launch_bounds__(32) void fgnn_proj_wmma(
    const float* __restrict__ vf,     // [B, 120, 128]
    const float* __restrict__ Wedge,  // [256, 16] row-major (rows 0..127 self, 128..255 nbr)
    float* __restrict__ proj)         // [B, 680, 32]
{
    __shared__ float Wl[32 * LSTRIDE];   // W^T staged: Wl[(tile*16+col)*LSTRIDE + k]

    const int lane  = threadIdx.x;       // 0..31 (wave32)
    const int b     = blockIdx.y;
    const int Mbase = blockIdx.x * 16;

    // Stage W transposed into LDS
    for (int idx = lane; idx < 256 * 16; idx += 32) {
        int kf = idx >> 4, col = idx & 15;
        int tile = kf >> 7, kl = kf & 127;
        Wl[(tile * 16 + col) * LSTRIDE + kl] = Wedge[idx];
    }
    __syncthreads();

    // Per-lane: which nff row, and its (up to 3) source rows in vf.
    // Direct rows use weights (1, 0); factor rows use (1/3, 1/3) => exact for
    // variable nodes, mean-of-3 for h/g factor rows.
    int n = Mbase + (lane & 15);
    if (n > NG - 1) n = NG - 1;          // pad rows compute garbage, not stored
    int s0, s1, s2; float sc, w;
    if (n < NVAR) {
        s0 = s1 = s2 = n; sc = 1.0f; w = 0.0f;
    } else if (n < GBASE) {              // h-factor: mean(vf[u], vf[v], vf[z_uv])
        int p = n - HBASE, u, v; decode_pair(p, u, v);
        s0 = u; s1 = v; s2 = ZBASE + p; sc = 1.0f / 3.0f; w = 1.0f / 3.0f;
    } else {                             // g-factor: mean of 3 z-nodes
        int t = n - GBASE, i, j, k; decode_tri(t, i, j, k);
        s0 = ZBASE + pair_rank(i, j);
        s1 = ZBASE + pair_rank(i, k);
        s2 = ZBASE + pair_rank(j, k);
        sc = 1.0f / 3.0f; w = 1.0f / 3.0f;
    }
    const float* r0 = vf + ((size_t)b * NVAR + s0) * DIM;
    const float* r1 = vf + ((size_t)b * NVAR + s1) * DIM;
    const float* r2 = vf + ((size_t)b * NVAR + s2) * DIM;
    __builtin_prefetch(r0, 0, 1);        // global_prefetch_b8
    __builtin_prefetch(r1, 0, 1);
    __builtin_prefetch(r2, 0, 1);

    // A-matrix 16x4 f32 layout: lanes 0-15 hold K={0,1}, lanes 16-31 hold K={2,3}
    const int kk  = (lane >> 4) << 1;    // 0 or 2
    const int col = lane & 15;
    const float* bs0 = &Wl[col * LSTRIDE];
    const float* bs1 = &Wl[(16 + col) * LSTRIDE];

    v8f c0 = {};
    v8f c1 = {};
    #pragma unroll 4
    for (int k0 = 0; k0 < DIM; k0 += 4) {
        const int kc = k0 + kk;
        float2 l0 = *(const float2*)(r0 + kc);
        float2 l1 = *(const float2*)(r1 + kc);
        float2 l2 = *(const float2*)(r2 + kc);
        v2f a;
        a.x = fmaf(l0.x, sc, (l1.x + l2.x) * w);
        a.y = fmaf(l0.y, sc, (l1.y + l2.y) * w);
        v2f bv0 = *(const v2f*)(bs0 + kc);
        v2f bv1 = *(const v2f*)(bs1 + kc);
        // D = A*B + C   (exact f32 matrix path)
        c0 = __builtin_amdgcn_wmma_f32_16x16x4_f32(false, a, false, bv0,
                                                   (short)0, c0, false, false);
        c1 = __builtin_amdgcn_wmma_f32_16x16x4_f32(false, a, false, bv1,
                                                   (short)0, c1, false, false);
    }

    // C/D layout: VGPR j -> row j (lanes 0-15) or row j+8 (lanes 16-31), col = lane&15
    const int rowBase = Mbase + ((lane >> 4) << 3);
    #pragma unroll
    for (int j = 0; j < 8; ++j) {
        int row = rowBase + j;
        if (row < NG) {
            float* p = proj + ((size_t)b * NG + row) * 32;
            p[col]      = c0[j];   // self-projection  (nff @ W_self)
            p[16 + col] = c1[j];   // neighbor-projection (nff @ W_nbr)
        }
    }
}

// =====================================================================
// Kernel B: per-edge relu + mean over 14 incident edges.
// Neighbor lists derived in closed form (matches reference graph incl. pads).
// =====================================================================
__global__ __launch_bounds__(128) void fgnn_edge_agg(
    const float* __restrict__ proj,   // [B, 680, 32]
    const float* __restrict__ bedge,  // [16]
    float* __restrict__ out)          // [B, 680, 16]
{
    const int tid = threadIdx.x;
    const int n   = blockIdx.x * 8 + (tid >> 4);
    const int e   = tid & 15;
    const int b   = blockIdx.y;
    if (n >= NG) return;

    int nb[DEG];
    if (n < NPER) {
        // person p: 14 h-factors containing p, in pair-rank order
        #pragma unroll
        for (int d = 0; d < DEG; ++d) {
            int u, v;
            if (d < n) { u = d; v = n; } else { u = n; v = d + 1; }
            nb[d] = HBASE + pair_rank(u, v);
        }
    } else if (n < NVAR) {
        // z-node (u,v): its h-factor, then 13 g-factors (third vertex ascending)
        int p = n - ZBASE, u, v; decode_pair(p, u, v);
        nb[0] = HBASE + p;
        for (int d = 1; d < DEG; ++d) {
            int w = d - 1;
            if (w >= u) ++w;
            if (w >= v) ++w;
            int i, j, k;
            if (w < u)      { i = w; j = u; k = v; }
            else if (w < v) { i = u; j = w; k = v; }
            else            { i = u; j = v; k = w; }
            nb[d] = GBASE + tri_rank(i, j, k);
        }
    } else if (n < GBASE) {
        // h-factor (u,v): [u, v, z_uv x12]  (reference pads with last element)
        int p = n - HBASE, u, v; decode_pair(p, u, v);
        nb[0] = u; nb[1] = v;
        for (int d = 2; d < DEG; ++d) nb[d] = ZBASE + p;
    } else {
        // g-factor (i,j,k): [z_ij, z_ik, z_jk x12]
        int t = n - GBASE, i, j, k; decode_tri(t, i, j, k);
        nb[0] = ZBASE + pair_rank(i, j);
        nb[1] = ZBASE + pair_rank(i, k);
        int z3 = ZBASE + pair_rank(j, k);
        for (int d = 2; d < DEG; ++d) nb[d] = z3;
    }

    const float* pb = proj + (size_t)b * NG * 32;
    const float selfv = pb[n * 32 + e] + bedge[e];
    float acc = 0.0f;
    #pragma unroll
    for (int d = 0; d < DEG; ++d) {
        float x = selfv + pb[nb[d] * 32 + 16 + e];
        acc += fmaxf(x, 0.0f);
    }
    out[((size_t)b * NG + n) * EDIM + e] = acc * (1.0f / 14.0f);
}

// =====================================================================
extern "C" void kernel_launch(void* const* d_in, const int* in_sizes, int n_in,
                              void* d_out, int out_size, void* d_ws, size_t ws_size,
                              hipStream_t stream) {
    const float* node_feats = (const float*)d_in[0];  // [B,120,128] f32
    const float* W_edge     = (const float*)d_in[1];  // [256,16]    f32
    const float* b_edge     = (const float*)d_in[2];  // [16]        f32
    float* out  = (float*)d_out;                      // [B,680,16]  f32
    float* proj = (float*)d_ws;                       // [B,680,32]  f32 (5.6 MB)

    const int B = in_sizes[0] / (NVAR * DIM);         // 64

    dim3 gridA((NG + 15) / 16, B);                    // 43 x 64 waves
    fgnn_proj_wmma<<<gridA, 32, 0, stream>>>(node_feats, W_edge, proj);

    dim3 gridB((NG + 7) / 8, B);                      // 85 x 64 blocks
    fgnn_edge_agg<<<gridB, 128, 0, stream>>>(proj, b_edge, out);
}